// AttSTWNBlock_8641474200437
// MI455X (gfx1250) — compile-verified
//
#include <hip/hip_runtime.h>
#include <math.h>

// Problem constants (from reference)
#define BB   2
#define CIN  32
#define COUT 64
#define NN   4096
#define TT   32
#define KS   3
#define EPSV 5e-5f

typedef __attribute__((ext_vector_type(2))) float v2f;
typedef __attribute__((ext_vector_type(8))) float v8f;

// ---------------------------------------------------------------------------
// Kernel 1: transpose/pack  x[b][f][s][t]  ->  xT[b][t][f][s]
// ---------------------------------------------------------------------------
__global__ __launch_bounds__(256) void xpose_kernel(const float* __restrict__ x,
                                                    float* __restrict__ xT) {
  __shared__ float tile[32][33];
  const int bf = blockIdx.x;        // b*CIN + f
  const int s0 = blockIdx.y * 32;
  const int b  = bf >> 5;
  const int f  = bf & 31;
  const int tx = threadIdx.x;       // 0..31
  const int ty = threadIdx.y;       // 0..7
#pragma unroll
  for (int i = 0; i < 4; ++i) {
    int s = ty + i * 8;
    tile[s][tx] = __builtin_nontemporal_load(&x[(bf * NN + s0 + s) * TT + tx]);
  }
  __syncthreads();
#pragma unroll
  for (int i = 0; i < 4; ++i) {
    int t = ty + i * 8;
    xT[((b * TT + t) * CIN + f) * NN + s0 + tx] = tile[tx][t];
  }
}

// ---------------------------------------------------------------------------
// Kernel 2: fused GEMM (WMMA f32) + upsample GEMM (WMMA) + attention epilogue.
// Double-buffered staging via GLOBAL_LOAD_ASYNC_TO_LDS_B128 (ASYNCcnt).
//
// Grid: blockIdx.x = bt (64, fast), blockIdx.y = n-block (64) -- concurrent
// blocks share one 3 MB W panel; xT (32 MB) stays L2-resident for the grid.
//
// Dynamic LDS layout (bytes):
//   [     0,  24576)  Upk : packed float2 [k][fpair16][o64]   (persistent)
//   [ 24576,  81920)  stage0 : A0 [32][16 float4, xor-swizzled] (8192B) @24576
//                              B0 f32[rowpair32][k3][2][64]   (49152B) @32768
//   [ 81920, 139264)  stage1 : A1 @81920, B1 @90112
//   epilogue reuse: Ypk float2[k][16][64] @24576 (24576B)
//                   Zf  f32[k][64][64]    @49152 (49152B)
//                   Wg  f32[3][64]        @98304 (768B)
// ---------------------------------------------------------------------------
#define SMEM_BYTES 139264
#define STAGE_STRIDE 57344

__global__ __launch_bounds__(256) void fused_kernel(
    const float* __restrict__ xT,
    const float* __restrict__ wav,    // (K, N, N)  wavelets[k][s][n]
    const float* __restrict__ ups,    // (K, CIN, COUT)
    const float* __restrict__ attu,   // (K, COUT)
    float* __restrict__ out) {        // (B, COUT, N, T)
  extern __shared__ char smem[];
  v2f*   UpP = (v2f*)smem;                    // 3*1024 float2
  v2f*   YpP = (v2f*)(smem + 24576);          // 3*1024 float2 (epilogue)
  float* Zf  = (float*)(smem + 49152);        // 3*4096 f32    (epilogue)
  float* Wg  = (float*)(smem + 98304);        // 3*64   f32    (epilogue)

  const int tid  = threadIdx.x;
  const int lane = tid & 31;
  const int wave = tid >> 5;                  // 8 waves
  const int lo   = lane & 15;
  const int hi   = lane >> 4;                 // 0 or 1

  const int bt = blockIdx.x;                  // 0..63  (fast dim: share W panel)
  const int n0 = blockIdx.y * 64;             // n tile
  const int b  = bt >> 5;
  const int t  = bt & 31;
  const int c0 = bt * 32;                     // xT row base: rows = channels f

  // ---- stage U packed as K-pairs: Up[k][f>>1][o].{x,y} = U[k][f][o]
  for (int idx = tid; idx < KS * CIN * COUT; idx += 256) {
    int k   = idx / (CIN * COUT);
    int rem = idx - k * (CIN * COUT);
    int f   = rem >> 6;
    int o   = rem & 63;
    ((float*)UpP)[(k * 1024 + (f >> 1) * 64 + o) * 2 + (f & 1)] = ups[idx];
  }

  // Issue one stage of async global->LDS copies: 14 instructions per wave.
  auto issue_stage = [&](int buf, int s0) {
    unsigned abase = 24576u + (unsigned)buf * STAGE_STRIDE;
    unsigned bbase = abase + 8192u;
    // A tile: 32 rows x 16 float4, xor-swizzled (q ^ (row&15)).
#pragma unroll
    for (int it = 0; it < 2; ++it) {
      int idx = it * 256 + tid;               // 0..511
      int row = idx >> 4;
      int q   = idx & 15;
      const float* g = &xT[(size_t)(c0 + row) * NN + s0 + (q << 2)];
      unsigned l = abase + (unsigned)(((row << 4) + (q ^ (row & 15))) << 4);
      asm volatile("global_load_async_to_lds_b128 %0, %1, off"
                   :: "v"(l), "v"(g) : "memory");
    }
    // B tiles: pair-grouped layout B[rowpair][k][rowparity][64] so the two
    // rows of a WMMA fragment are the nearest (256B) mergeable LDS pair.
#pragma unroll
    for (int k = 0; k < KS; ++k) {
      const float* wk = wav + (size_t)k * NN * NN;
#pragma unroll
      for (int it = 0; it < 4; ++it) {
        int idx = it * 256 + tid;             // 0..1023
        int row = idx >> 4;                   // s_local 0..63
        int c4  = (idx & 15) << 2;            // n_local
        const float* g = &wk[(size_t)(s0 + row) * NN + n0 + c4];
        int grp = (row >> 1) * 6 + k * 2 + (row & 1);
        unsigned l = bbase + (unsigned)(((grp << 6) + c4) << 2);
        asm volatile("global_load_async_to_lds_b128 %0, %1, off"
                     :: "v"(l), "v"(g) : "memory");
      }
    }
  };

  // per-wave output tile of the main GEMM
  const int mi    = wave & 1;                 // M subtile (2 x 16 rows = f)
  const int ni    = wave >> 1;                // N subtile (4 x 16 cols = n)
  const int m_row = mi * 16 + lo;
  const int n_col = ni * 16 + lo;
  v8f acc0 = {}, acc1 = {}, acc2 = {};

  // ====================== main GEMM loop (double-buffered) =================
  issue_stage(0, 0);
  for (int step = 0; step < NN / 64; ++step) {
    const int s0  = step * 64;
    const int buf = step & 1;
    const bool has_next = (step + 1) < (NN / 64);
    if (has_next) {
      issue_stage(buf ^ 1, s0 + 64);
      asm volatile("s_wait_asynccnt 14" ::: "memory");  // current stage landed
    } else {
      asm volatile("s_wait_asynccnt 0" ::: "memory");
    }
    __syncthreads();

    const char*  Ab = smem + 24576 + buf * STAGE_STRIDE;
    const float* Bb = (const float*)(smem + 32768 + buf * STAGE_STRIDE);
#pragma unroll 4
    for (int kk = 0; kk < 64; kk += 4) {
      // A fragment: pair (kc,kc+1) of row m_row; xor-swizzled float4 layout.
      int qq = kk >> 2;
      v2f a  = *(const v2f*)(Ab + ((((m_row << 4) + (qq ^ (m_row & 15))) << 4)
                                   + (hi << 3)));
      int rp = (kk >> 1) + hi;                // row-pair index (kc even)
      const float* Br = Bb + ((rp * 6) << 6) + n_col;
      v2f b0; b0.x = Br[0];    b0.y = Br[64];           // k=0: rows kc, kc+1
      acc0 = __builtin_amdgcn_wmma_f32_16x16x4_f32(false, a, false, b0,
                                                   (short)0, acc0, false, false);
      v2f b1; b1.x = Br[128];  b1.y = Br[192];          // k=1
      acc1 = __builtin_amdgcn_wmma_f32_16x16x4_f32(false, a, false, b1,
                                                   (short)0, acc1, false, false);
      v2f b2; b2.x = Br[256];  b2.y = Br[320];          // k=2
      acc2 = __builtin_amdgcn_wmma_f32_16x16x4_f32(false, a, false, b2,
                                                   (short)0, acc2, false, false);
    }
    __syncthreads();  // all waves done with buf before it is refilled
  }

  // ==================== epilogue: write Y K-pair packed ====================
  {
    float* yf = (float*)YpP;
#pragma unroll
    for (int r = 0; r < 8; ++r) {
      int f    = mi * 16 + r + 8 * hi;        // C/D layout: M = r + 8*hi
      int base = ((f >> 1) * 64 + n_col) * 2 + (f & 1);
      yf[0 * 2048 + base] = acc0[r];
      yf[1 * 2048 + base] = acc1[r];
      yf[2 * 2048 + base] = acc2[r];
    }
  }
  __syncthreads();

  // ================= Z_k(64x64) = U_k^T(64x32) @ Y_k(32x64) ===============
  for (int p = wave; p < 16; p += 8) {        // 2 tile positions per wave
    int oi     = p & 3;
    int ni2    = p >> 2;
    int o_col  = oi * 16 + lo;
    int n_col2 = ni2 * 16 + lo;
    v8f z0 = {}, z1 = {}, z2 = {};
#pragma unroll
    for (int kk = 0; kk < 32; kk += 4) {
      int fp = (kk >> 1) + hi;
      v2f ua0 = *(v2f*)(UpP + 0 * 1024 + fp * 64 + o_col);
      v2f ya0 = *(v2f*)(YpP + 0 * 1024 + fp * 64 + n_col2);
      z0 = __builtin_amdgcn_wmma_f32_16x16x4_f32(false, ua0, false, ya0,
                                                 (short)0, z0, false, false);
      v2f ua1 = *(v2f*)(UpP + 1 * 1024 + fp * 64 + o_col);
      v2f ya1 = *(v2f*)(YpP + 1 * 1024 + fp * 64 + n_col2);
      z1 = __builtin_amdgcn_wmma_f32_16x16x4_f32(false, ua1, false, ya1,
                                                 (short)0, z1, false, false);
      v2f ua2 = *(v2f*)(UpP + 2 * 1024 + fp * 64 + o_col);
      v2f ya2 = *(v2f*)(YpP + 2 * 1024 + fp * 64 + n_col2);
      z2 = __builtin_amdgcn_wmma_f32_16x16x4_f32(false, ua2, false, ya2,
                                                 (short)0, z2, false, false);
    }
#pragma unroll
    for (int r = 0; r < 8; ++r) {
      int o = oi * 16 + r + 8 * hi;
      Zf[0 * 4096 + o * 64 + n_col2] = z0[r];
      Zf[1 * 4096 + o * 64 + n_col2] = z1[r];
      Zf[2 * 4096 + o * 64 + n_col2] = z2[r];
    }
  }
  __syncthreads();

  // ============== attention: score, normalize (ddof=1), softmax ============
  if (tid < 64) {
    int n = tid;
    float av[3];
#pragma unroll
    for (int k = 0; k < 3; ++k) {
      float s = 0.f;
      for (int o = 0; o < 64; ++o)
        s += Zf[k * 4096 + o * 64 + n] * attu[k * 64 + o];
      av[k] = s;
    }
    float mu = (av[0] + av[1] + av[2]) * (1.f / 3.f);
    float d0 = av[0] - mu, d1 = av[1] - mu, d2 = av[2] - mu;
    float sd  = sqrtf((d0 * d0 + d1 * d1 + d2 * d2) * 0.5f);  // unbiased, K-1=2
    float inv = 1.f / (sd + EPSV);
    float e0 = __expf(d0 * inv), e1 = __expf(d1 * inv), e2 = __expf(d2 * inv);
    float es = 1.f / (e0 + e1 + e2);
    Wg[0 * 64 + n] = e0 * es;
    Wg[1 * 64 + n] = e1 * es;
    Wg[2 * 64 + n] = e2 * es;
  }
  __syncthreads();

  // ===== weighted combine, store out[b][o][n][t] (NT: don't pollute L2) ====
  for (int idx = tid; idx < COUT * 64; idx += 256) {
    int n = idx & 63;
    int o = idx >> 6;
    float r = Wg[n]       * Zf[o * 64 + n]
            + Wg[64 + n]  * Zf[4096 + o * 64 + n]
            + Wg[128 + n] * Zf[8192 + o * 64 + n];
    __builtin_nontemporal_store(r,
        &out[(((size_t)b * COUT + o) * NN + n0 + n) * TT + t]);
  }
}

// ---------------------------------------------------------------------------
extern "C" void kernel_launch(void* const* d_in, const int* in_sizes, int n_in,
                              void* d_out, int out_size, void* d_ws, size_t ws_size,
                              hipStream_t stream) {
  const float* x    = (const float*)d_in[0];   // (B, CIN, N, T)
  const float* wav  = (const float*)d_in[1];   // (K, N, N)
  const float* ups  = (const float*)d_in[2];   // (K, CIN, COUT)
  const float* attu = (const float*)d_in[3];   // (K, COUT)
  float* out = (float*)d_out;
  float* xT  = (float*)d_ws;                   // 2*32*32*4096*4 = 32 MiB

  dim3 g1(BB * CIN, NN / 32), b1(32, 8);
  xpose_kernel<<<g1, b1, 0, stream>>>(x, xT);

  dim3 g2(BB * TT, NN / 64), b2(256);
  fused_kernel<<<g2, b2, SMEM_BYTES, stream>>>(xT, wav, ups, attu, out);
}